// GConvLSTM_Simple_38173669327258
// MI455X (gfx1250) — compile-verified
//
#include <hip/hip_runtime.h>
#include <hip/hip_bf16.h>
#include <math.h>

typedef __attribute__((ext_vector_type(2))) float v2f;
typedef __attribute__((ext_vector_type(8))) float v8f;

#define CH 32  // CIN == COUT == 32

// ---------------------------------------------------------------------------
// K1: degree accumulation: deg[dst[e]] += w[e]   (hardware f32 atomics in L2)
// ---------------------------------------------------------------------------
__global__ __launch_bounds__(256) void k_deg(const int* __restrict__ dst,
                                             const float* __restrict__ w,
                                             float* __restrict__ deg, int E) {
    int e = blockIdx.x * blockDim.x + threadIdx.x;
    if (e < E) {
        __hip_atomic_fetch_add(&deg[dst[e]], w[e],
                               __ATOMIC_RELAXED, __HIP_MEMORY_SCOPE_AGENT);
    }
}

// ---------------------------------------------------------------------------
// K2: dinv[i] = deg>0 ? rsqrt(max(deg,1e-12)) : 0
// ---------------------------------------------------------------------------
__global__ __launch_bounds__(256) void k_dinv(const float* __restrict__ deg,
                                              float* __restrict__ dinv, int N) {
    int i = blockIdx.x * blockDim.x + threadIdx.x;
    if (i < N) {
        float d = deg[i];
        dinv[i] = (d > 0.0f) ? rsqrtf(fmaxf(d, 1e-12f)) : 0.0f;
    }
}

// ---------------------------------------------------------------------------
// K3: P = X@Wx + H@Wh  via V_WMMA_F32_16X16X4_F32  (fp32-exact)
// One wave handles a 16-row tile x 32 cols (two 16x16 D tiles).
// A 16x4 layout: lane(l<16)=row M=l, v0:K=2*half, v1:K=2*half+1
// B 4x16 layout: v0 = rows K={2*half}, v1 = rows K={2*half+1}, col = n0+lm
// D 16x16:      vgpr r -> row r + 8*half, col = lm
// ---------------------------------------------------------------------------
__global__ __launch_bounds__(256) void k_gemm_wmma(const float* __restrict__ X,
                                                   const float* __restrict__ H,
                                                   const float* __restrict__ Wx,
                                                   const float* __restrict__ Wh,
                                                   float* __restrict__ P, int N) {
    int wave = blockIdx.x * (blockDim.x >> 5) + (threadIdx.x >> 5);
    int row0 = wave * 16;
    if (row0 >= N) return;                 // wave-uniform: EXEC stays all-ones

    int lane = threadIdx.x & 31;
    int half = lane >> 4;
    int lm   = lane & 15;

    int arow = row0 + lm;
    if (arow > N - 1) arow = N - 1;        // clamp loads on (impossible here) tail
    const float* xrow = X + (size_t)arow * CH;
    const float* hrow = H + (size_t)arow * CH;

    v8f acc0 = {};
    v8f acc1 = {};

#pragma unroll
    for (int kt = 0; kt < 8; ++kt) {
        int k0 = kt * 4;
        int ka = k0 + 2 * half;

        v2f ax = { xrow[ka], xrow[ka + 1] };
        v2f ah = { hrow[ka], hrow[ka + 1] };

        int krow = ka * CH;
        v2f bx0 = { Wx[krow + lm],      Wx[krow + CH + lm]      };
        v2f bx1 = { Wx[krow + 16 + lm], Wx[krow + CH + 16 + lm] };
        v2f bh0 = { Wh[krow + lm],      Wh[krow + CH + lm]      };
        v2f bh1 = { Wh[krow + 16 + lm], Wh[krow + CH + 16 + lm] };

        acc0 = __builtin_amdgcn_wmma_f32_16x16x4_f32(false, ax, false, bx0,
                                                     (short)0, acc0, false, false);
        acc1 = __builtin_amdgcn_wmma_f32_16x16x4_f32(false, ax, false, bx1,
                                                     (short)0, acc1, false, false);
        acc0 = __builtin_amdgcn_wmma_f32_16x16x4_f32(false, ah, false, bh0,
                                                     (short)0, acc0, false, false);
        acc1 = __builtin_amdgcn_wmma_f32_16x16x4_f32(false, ah, false, bh1,
                                                     (short)0, acc1, false, false);
    }

    float* prow = P + (size_t)row0 * CH;
#pragma unroll
    for (int r = 0; r < 8; ++r) {
        int rr = r + 8 * half;
        if (row0 + rr < N) {
            prow[(size_t)rr * CH + lm]      = acc0[r];
            prow[(size_t)rr * CH + 16 + lm] = acc1[r];
        }
    }
}

// ---------------------------------------------------------------------------
// K4: edge scatter. 1 wave per edge, lane = channel.
//   agg[dst][c] += dinv[src]*w*dinv[dst] * P[src][c]
// Coalesced 128B gather + 32 hardware global_atomic_add_f32 (L2-resident).
// ---------------------------------------------------------------------------
__global__ __launch_bounds__(256) void k_scatter(const int* __restrict__ src,
                                                 const int* __restrict__ dst,
                                                 const float* __restrict__ w,
                                                 const float* __restrict__ dinv,
                                                 const float* __restrict__ P,
                                                 float* __restrict__ agg, int E) {
    int e = blockIdx.x * (blockDim.x >> 5) + (threadIdx.x >> 5);
    if (e >= E) return;
    int lane = threadIdx.x & 31;

    int s = src[e];
    int d = dst[e];
    float nrm = dinv[s] * w[e] * dinv[d];
    float val = nrm * P[(size_t)s * CH + lane];
    __hip_atomic_fetch_add(&agg[(size_t)d * CH + lane], val,
                           __ATOMIC_RELAXED, __HIP_MEMORY_SCOPE_AGENT);
}

// ---------------------------------------------------------------------------
// K5: LSTM gates. S = agg + bx + bh (shared across the 4 gates).
// out = [O | H_new | C_new], each N*32 floats.
// ---------------------------------------------------------------------------
__device__ __forceinline__ float sigmoidf_(float x) {
    return 1.0f / (1.0f + __expf(-x));
}

__global__ __launch_bounds__(256) void k_gates(const float* __restrict__ agg,
                                               const float* __restrict__ Cst,
                                               const float* __restrict__ bx,
                                               const float* __restrict__ bh,
                                               const float* __restrict__ bi,
                                               const float* __restrict__ bf,
                                               const float* __restrict__ bc,
                                               const float* __restrict__ bo,
                                               float* __restrict__ out, int NC) {
    int t = blockIdx.x * blockDim.x + threadIdx.x;
    if (t >= NC) return;
    int c = t & (CH - 1);

    float s  = agg[t] + bx[c] + bh[c];
    float I  = sigmoidf_(s + bi[c]);
    float F  = sigmoidf_(s + bf[c]);
    float T  = tanhf(s + bc[c]);
    float Cn = F * Cst[t] + I * T;
    float O  = sigmoidf_(s + bo[c]);
    float Hn = O * tanhf(Cn);

    out[t]          = O;
    out[NC + t]     = Hn;
    out[2 * NC + t] = Cn;
}

// ---------------------------------------------------------------------------
extern "C" void kernel_launch(void* const* d_in, const int* in_sizes, int n_in,
                              void* d_out, int out_size, void* d_ws, size_t ws_size,
                              hipStream_t stream) {
    const float* X   = (const float*)d_in[0];
    const int*   ei  = (const int*)  d_in[1];
    const float* w   = (const float*)d_in[2];
    const float* H   = (const float*)d_in[3];
    const float* C   = (const float*)d_in[4];
    const float* Wx  = (const float*)d_in[5];
    const float* bx  = (const float*)d_in[6];
    const float* Wh  = (const float*)d_in[7];
    const float* bh  = (const float*)d_in[8];
    const float* bi  = (const float*)d_in[9];
    const float* bf  = (const float*)d_in[10];
    const float* bc  = (const float*)d_in[11];
    const float* bo  = (const float*)d_in[12];

    const int N = in_sizes[0] / CH;     // 100000
    const int E = in_sizes[2];          // 3200000
    const int* src = ei;                // edge_index row 0
    const int* dst = ei + E;            // edge_index row 1

    // Workspace layout (floats): deg[Na] | dinv[Na] | P[N*32] | agg[N*32]
    float* ws  = (float*)d_ws;
    size_t Na  = ((size_t)N + 3) & ~(size_t)3;
    float* deg  = ws;
    float* dinv = ws + Na;
    float* P    = ws + 2 * Na;
    float* agg  = P + (size_t)N * CH;

    // Zero accumulators every call (harness does not re-poison between replays).
    hipMemsetAsync(deg, 0, (size_t)N * sizeof(float), stream);
    hipMemsetAsync(agg, 0, (size_t)N * CH * sizeof(float), stream);

    k_deg<<<(E + 255) / 256, 256, 0, stream>>>(dst, w, deg, E);
    k_dinv<<<(N + 255) / 256, 256, 0, stream>>>(deg, dinv, N);

    int tiles = (N + 15) / 16;                       // 16 rows per wave
    k_gemm_wmma<<<(tiles + 7) / 8, 256, 0, stream>>>(X, H, Wx, Wh, P, N);

    k_scatter<<<(E + 7) / 8, 256, 0, stream>>>(src, dst, w, dinv, P, agg, E);

    int NC = N * CH;
    k_gates<<<(NC + 255) / 256, 256, 0, stream>>>(agg, C, bx, bh, bi, bf, bc, bo,
                                                  (float*)d_out, NC);
}